// DTWDistance_27917287424508
// MI455X (gfx1250) — compile-verified
//
#include <hip/hip_runtime.h>
#include <math.h>

typedef __attribute__((ext_vector_type(2))) float v2f;
typedef __attribute__((ext_vector_type(8))) float v8f;

#define FDIM 128
#define TDIM 512
#define ROWS 16
#define PWAVES 4                     // producer (WMMA) waves
#define NWAVES (PWAVES + 1)          // + 1 DP consumer wave
#define TPW ((TDIM / 16) / PWAVES)   // 8 column tiles per producer wave
#define NBLK (TDIM / ROWS)           // 32 row blocks

__device__ __forceinline__ void produce_block(const float* __restrict__ xb,
                                              const float* __restrict__ yb,
                                              const float* __restrict__ sX2,
                                              const float* __restrict__ sY2,
                                              float* __restrict__ cbuf,
                                              int i0, int wave, int lane) {
    const int m16   = lane & 15;
    const int khalf = (lane >> 4) * 2;

    v8f acc[TPW];
    #pragma unroll
    for (int t = 0; t < TPW; ++t)
        #pragma unroll
        for (int r = 0; r < 8; ++r) acc[t][r] = 0.f;

    for (int k0 = 0; k0 < FDIM; k0 += 4) {
        v2f a;  // A[m][k] = x[b, k, i0+m]
        a.x = xb[(k0 + khalf) * TDIM + (i0 + m16)];
        a.y = xb[(k0 + khalf + 1) * TDIM + (i0 + m16)];
        #pragma unroll
        for (int t = 0; t < TPW; ++t) {
            const int n0 = (wave * TPW + t) * 16;
            v2f bf; // B[k][n] = y[b, k, n0+n]
            bf.x = yb[(k0 + khalf) * TDIM + (n0 + m16)];
            bf.y = yb[(k0 + khalf + 1) * TDIM + (n0 + m16)];
            acc[t] = __builtin_amdgcn_wmma_f32_16x16x4_f32(
                false, a, false, bf, (short)0, acc[t], false, false);
        }
    }
    // cost = sqrt(max(x2 + y2 - 2*xy, 0)) -> LDS (raw v_sqrt_f32: inputs are
    // normal-range distances, no denorm/overflow fixup needed)
    #pragma unroll
    for (int t = 0; t < TPW; ++t) {
        const int n0 = (wave * TPW + t) * 16;
        #pragma unroll
        for (int r = 0; r < 8; ++r) {
            const int m = (lane < 16) ? r : (r + 8);
            const int n = n0 + m16;
            float d2 = sX2[i0 + m] + sY2[n] - 2.0f * acc[t][r];
            cbuf[m * TDIM + n] = __builtin_amdgcn_sqrtf(fmaxf(d2, 0.f));
        }
    }
}

__global__ __launch_bounds__(NWAVES * 32)
void dtw_fused_kernel(const float* __restrict__ X,
                      const float* __restrict__ Y,
                      float* __restrict__ out) {
    __shared__ float sX2[TDIM];
    __shared__ float sY2[TDIM];
    __shared__ float sCost[2][ROWS * TDIM];   // 64 KB double buffer (320 KB WGP LDS)

    const int b    = blockIdx.x;
    const int tid  = threadIdx.x;
    const int lane = tid & 31;
    const int wave = tid >> 5;
    const float INF = __builtin_inff();

    const float* xb = X + (size_t)b * FDIM * TDIM;
    const float* yb = Y + (size_t)b * FDIM * TDIM;

    // ---- Phase 0: column squared norms (warms L2 for the GEMM) ----
    for (int t = tid; t < TDIM; t += blockDim.x) {
        float sx = 0.f, sy = 0.f;
        for (int f = 0; f < FDIM; ++f) {
            float xv = xb[f * TDIM + t];
            float yv = yb[f * TDIM + t];
            sx += xv * xv;
            sy += yv * yv;
        }
        sX2[t] = sx;
        sY2[t] = sy;
    }
    __syncthreads();

    // ---- Prologue: producers fill buffer 0 ----
    if (wave < PWAVES)
        produce_block(xb, yb, sX2, sY2, sCost[0], 0, wave, lane);
    __syncthreads();

    // DP row state lives in wave-4 registers: preg[k] = dtw_prev[lane*16 + k]
    float preg[16];
    #pragma unroll
    for (int k = 0; k < 16; ++k) preg[k] = INF;
    float leftB = 0.f;   // dtw[-1][-1] boundary: 0 for first row, INF afterwards

    // ---- Software pipeline: WMMA(blk+1) overlaps DP(blk) ----
    for (int blk = 0; blk < NBLK; ++blk) {
        if (wave < PWAVES) {
            if (blk + 1 < NBLK)
                produce_block(xb, yb, sX2, sY2, sCost[(blk + 1) & 1],
                              (blk + 1) * ROWS, wave, lane);
        } else {
            const float* cbuf = sCost[blk & 1];
            for (int r = 0; r < ROWS; ++r) {
                // cost segment for this lane: 16 contiguous floats -> 4x ds_load_b128
                const float4* crow4 = (const float4*)(cbuf + r * TDIM + lane * 16);
                float cseg[16];
                #pragma unroll
                for (int q = 0; q < 4; ++q) {
                    float4 v = crow4[q];
                    cseg[4 * q + 0] = v.x; cseg[4 * q + 1] = v.y;
                    cseg[4 * q + 2] = v.z; cseg[4 * q + 3] = v.w;
                }
                // P[j-1] at segment start comes from the previous lane
                float plast = __shfl_up(preg[15], 1, 32);
                if (lane == 0) plast = leftB;

                // pass 1: lane-local composition of f_j(x) = min(b_j, c_j + x)
                float bseg[16];
                float Bacc = INF, Cacc = 0.f;
                #pragma unroll
                for (int k = 0; k < 16; ++k) {
                    float pm1 = (k == 0) ? plast : preg[k - 1];
                    float e   = fminf(preg[k], pm1);     // min(up, diag)
                    float c   = cseg[k];
                    float bb  = c + e;
                    bseg[k]   = bb;
                    Bacc = fminf(bb, c + Bacc);
                    Cacc = c + Cacc;
                }
                // inclusive cross-lane scan of (B, C) in the (min,+) semiring
                #pragma unroll
                for (int off = 1; off < 32; off <<= 1) {
                    float Bo = __shfl_up(Bacc, off, 32);
                    float Co = __shfl_up(Cacc, off, 32);
                    if (lane >= off) {
                        Bacc = fminf(Bacc, Cacc + Bo);
                        Cacc = Cacc + Co;
                    }
                }
                float xin = __shfl_up(Bacc, 1, 32);   // cur at prev lane's last col
                if (lane == 0) xin = INF;
                // pass 2: regenerate per-column cur into the register row
                float x = xin;
                #pragma unroll
                for (int k = 0; k < 16; ++k) {
                    x = fminf(bseg[k], cseg[k] + x);
                    preg[k] = x;
                }
                leftB = INF;   // boundary closes after the first DP row
            }
        }
        __syncthreads();
    }

    // final DTW value = last row, last column = lane 31's preg[15]
    if (wave == PWAVES && lane == 31) out[b] = preg[15];
}

extern "C" void kernel_launch(void* const* d_in, const int* in_sizes, int n_in,
                              void* d_out, int out_size, void* d_ws, size_t ws_size,
                              hipStream_t stream) {
    const float* x = (const float*)d_in[0];
    const float* y = (const float*)d_in[1];
    float* out = (float*)d_out;
    const int B = out_size;   // 128
    dtw_fused_kernel<<<B, NWAVES * 32, 0, stream>>>(x, y, out);
}